// HyperCOMPASNN_28561532518519
// MI455X (gfx1250) — compile-verified
//
#include <hip/hip_runtime.h>
#include <hip/hip_bf16.h>

typedef __attribute__((ext_vector_type(2))) float v2f;
typedef __attribute__((ext_vector_type(8))) float v8f;

#define NEG_SLOPE 0.01f

// ---------------------------------------------------------------------------
// Kernel 1: serial trunk MLP. 1 block x 64 threads (2 waves), one row/lane.
// h broadcast between layers through LDS. Pure fp32 VALU: M=1 makes matrix
// ops >=94% idle, and this is a latency chain anyway.
// ---------------------------------------------------------------------------
__global__ void hyper_trunk_kernel(const float* __restrict__ x,
                                   const float* __restrict__ Win,
                                   const float* __restrict__ bin,
                                   const float* __restrict__ Ws,
                                   const float* __restrict__ bs,
                                   float* __restrict__ f_out) {
    __shared__ float hbuf[64];
    __shared__ float xs[16];
    const int t = threadIdx.x;  // 0..63

    if (t < 13) xs[t] = x[t];
    __syncthreads();

    // layer 0: h = x @ W_in.T + b_in   (no activation on input)
    float acc = bin[t];
#pragma unroll
    for (int k = 0; k < 13; ++k) acc += Win[t * 13 + k] * xs[k];
    hbuf[t] = acc;
    __syncthreads();

    // 20x: h = leaky_relu(h) @ W.T + b
    for (int L = 0; L < 20; ++L) {
        const float* W = Ws + (size_t)L * 64 * 64;
        float a = bs[L * 64 + t];
#pragma unroll
        for (int k = 0; k < 64; ++k) {
            float g = hbuf[k];
            g = (g > 0.0f) ? g : NEG_SLOPE * g;
            a += W[t * 64 + k] * g;
        }
        __syncthreads();
        hbuf[t] = a;
        __syncthreads();
    }
    f_out[t] = hbuf[t];  // final h, no trailing activation
}

// ---------------------------------------------------------------------------
// Kernel 2: heads. out[r] = dot(W[r,:], f) + b[r] for 7681 rows total.
// One wave per 16-row tile using V_WMMA_F32_16X16X4_F32:
//   A = 16x4 weight tile, B = f[4s..4s+3] broadcast over all 16 columns,
//   C preloaded with bias -> D row m (replicated over N) = output row.
// Operand layouts per CDNA5 ISA 7.12.2 (32-bit A/B: VGPR0={K0,K2},
// VGPR1={K1,K3} across lane halves; C/D VGPR r = rows r, r+8).
// ---------------------------------------------------------------------------
__global__ void hyper_heads_kernel(const float* __restrict__ f,
                                   const float* __restrict__ fc1wW, const float* __restrict__ fc1wB,
                                   const float* __restrict__ fc1bW, const float* __restrict__ fc1bB,
                                   const float* __restrict__ fc2wW, const float* __restrict__ fc2wB,
                                   const float* __restrict__ fc2bW, const float* __restrict__ fc2bB,
                                   const float* __restrict__ fc3wW, const float* __restrict__ fc3wB,
                                   const float* __restrict__ fc3bW, const float* __restrict__ fc3bB,
                                   const float* __restrict__ fc4wW, const float* __restrict__ fc4wB,
                                   const float* __restrict__ fc4bW, const float* __restrict__ fc4bB,
                                   float* __restrict__ out) {
    const int gtid = blockIdx.x * blockDim.x + threadIdx.x;
    const int wave = gtid >> 5;         // global wave id (wave32)
    const int lane = threadIdx.x & 31;

    if (wave > 480) return;

    if (wave == 480) {
        // fc4_bias: single row -> wave-level dot product + shuffle reduce.
        float s = f[lane] * fc4bW[lane] + f[lane + 32] * fc4bW[lane + 32];
        for (int off = 16; off > 0; off >>= 1) s += __shfl_xor(s, off, 32);
        if (lane == 0) out[7680] = s + fc4bB[0];
        return;
    }

    // Wave-uniform section lookup (tile counts: 84,4,256,4,128,2,2 = 480).
    const float* W;
    const float* B;
    int obase, tstart;
    if (wave < 84)       { W = fc1wW; B = fc1wB; obase = 0;    tstart = 0;   }
    else if (wave < 88)  { W = fc1bW; B = fc1bB; obase = 1344; tstart = 84;  }
    else if (wave < 344) { W = fc2wW; B = fc2wB; obase = 1408; tstart = 88;  }
    else if (wave < 348) { W = fc2bW; B = fc2bB; obase = 5504; tstart = 344; }
    else if (wave < 476) { W = fc3wW; B = fc3wB; obase = 5568; tstart = 348; }
    else if (wave < 478) { W = fc3bW; B = fc3bB; obase = 7616; tstart = 476; }
    else                 { W = fc4wW; B = fc4wB; obase = 7648; tstart = 478; }

    const int rbase = (wave - tstart) * 16;     // first row of this tile
    const int m     = lane & 15;                // row within tile
    const int koff  = (lane >> 4) << 1;         // 0 (lanes 0-15) / 2 (lanes 16-31)
    const int boff  = (lane >> 4) << 3;         // C/D row offset per lane half

    // B operands: f[4s+koff], f[4s+koff+1] -> {VGPR0, VGPR1}; broadcast over N.
    v2f bv[16];
#pragma unroll
    for (int s = 0; s < 16; ++s) {
        bv[s].x = f[4 * s + koff];
        bv[s].y = f[4 * s + koff + 1];
    }

    // C init = bias (replicated across N per the C/D layout).
    v8f acc;
#pragma unroll
    for (int r = 0; r < 8; ++r) acc[r] = B[rbase + boff + r];

    // K loop: 16 steps of K=4, one f32 WMMA each.
    const float* Wrow = W + (size_t)(rbase + m) * 64 + koff;
#pragma unroll
    for (int s = 0; s < 16; ++s) {
        v2f a;
        a.x = Wrow[4 * s];
        a.y = Wrow[4 * s + 1];
        acc = __builtin_amdgcn_wmma_f32_16x16x4_f32(
            /*neg_a=*/false, a, /*neg_b=*/false, bv[s],
            /*c_mod=*/(short)0, acc, /*reuse_a=*/false, /*reuse_b=*/false);
    }

    // D row m is replicated across lanes of its half; lanes 0 and 16 store 8 each.
    if ((lane & 15) == 0) {
        float* o = out + obase + rbase + boff;
#pragma unroll
        for (int r = 0; r < 8; ++r) o[r] = acc[r];
    }
}

// ---------------------------------------------------------------------------
extern "C" void kernel_launch(void* const* d_in, const int* in_sizes, int n_in,
                              void* d_out, int out_size, void* d_ws, size_t ws_size,
                              hipStream_t stream) {
    const float* context = (const float*)d_in[0];
    const float* Win     = (const float*)d_in[1];
    const float* bin     = (const float*)d_in[2];
    const float* Ws      = (const float*)d_in[3];
    const float* bs      = (const float*)d_in[4];
    const float* fc1wW   = (const float*)d_in[5];
    const float* fc1wB   = (const float*)d_in[6];
    const float* fc1bW   = (const float*)d_in[7];
    const float* fc1bB   = (const float*)d_in[8];
    const float* fc2wW   = (const float*)d_in[9];
    const float* fc2wB   = (const float*)d_in[10];
    const float* fc2bW   = (const float*)d_in[11];
    const float* fc2bB   = (const float*)d_in[12];
    const float* fc3wW   = (const float*)d_in[13];
    const float* fc3wB   = (const float*)d_in[14];
    const float* fc3bW   = (const float*)d_in[15];
    const float* fc3bB   = (const float*)d_in[16];
    const float* fc4wW   = (const float*)d_in[17];
    const float* fc4wB   = (const float*)d_in[18];
    const float* fc4bW   = (const float*)d_in[19];
    const float* fc4bB   = (const float*)d_in[20];

    float* out = (float*)d_out;
    float* f   = (float*)d_ws;  // 64-float trunk output

    hyper_trunk_kernel<<<1, 64, 0, stream>>>(context, Win, bin, Ws, bs, f);

    // 480 WMMA tiles + 1 scalar wave = 481 waves; 61 blocks x 8 waves = 488.
    hyper_heads_kernel<<<61, 256, 0, stream>>>(f,
        fc1wW, fc1wB, fc1bW, fc1bB,
        fc2wW, fc2wB, fc2bW, fc2bB,
        fc3wW, fc3wB, fc3bW, fc3bB,
        fc4wW, fc4wB, fc4bW, fc4bB,
        out);
}